// NeuralDict_16157666968039
// MI455X (gfx1250) — compile-verified
//
#include <hip/hip_runtime.h>
#include <float.h>
#include <math.h>

// Cosine-similarity retrieval: argmax_n <p_n,x>/||p_n||, output patterns[argmax].
// Bandwidth-bound (205 MB stream ~ 8.8us @ 23.3 TB/s). WMMA f32 16x16x4 folds
// 16 row-dots per wave tile; double-buffered async global->LDS b128 staging
// keeps HBM requests continuously in flight. Last stage is peeled so the
// steady-state loop is branch-free: issue(next) -> s_wait_asynccnt 4 -> compute.

typedef __attribute__((ext_vector_type(2))) float v2f;
typedef __attribute__((ext_vector_type(8))) float v8f;

#define DIM          512
#define WAVES_PB     8
#define TILE_ROWS    16
#define KCHUNK       32                   // columns staged per async stage
#define ROW_F        (KCHUNK + 4)         // padded LDS row: 36 floats = 144B (16B aligned; 36m mod 64 distinct -> no bank conflicts)
#define ROW_B        (ROW_F * 4)
#define TILE_B       (TILE_ROWS * ROW_B)  // 2304 B per stage buffer
#define X_B          (DIM * 4)            // 2048 B staged x
// LDS: 2048 + 8 waves * 2 buffers * 2304 = 38912 B

__global__ __launch_bounds__(256) void nd_phase1(const float* __restrict__ x,
                                                 const float* __restrict__ patterns,
                                                 int N, float* __restrict__ ws) {
  __shared__ __align__(16) unsigned char lds[X_B + WAVES_PB * 2 * TILE_B];
  __shared__ float swv[WAVES_PB];
  __shared__ int   swi[WAVES_PB];

  float* ldsx = (float*)lds;
  const int tid  = threadIdx.x;
  const int lane = tid & 31;
  const int wave = tid >> 5;

  // Stage x once per block (2KB, coalesced).
  for (int i = tid; i < DIM; i += 256) ldsx[i] = x[i];
  __syncthreads();

  const int tile = blockIdx.x * WAVES_PB + wave;
  const int row0 = tile * TILE_ROWS;

  float bestVal = -FLT_MAX;
  int   bestIdx = 0x7fffffff;

  if (row0 < N) {
    float* tl0 = (float*)(lds + X_B + wave * 2 * TILE_B);
    float* tl1 = tl0 + TILE_B / 4;
    const unsigned buf0 = (unsigned)(uintptr_t)tl0;   // low 32 bits = LDS byte addr
    const unsigned buf1 = buf0 + TILE_B;
    const int m     = lane & 15;   // A-matrix row handled by this lane
    const int khalf = lane >> 4;   // 0: K pair {0,1}; 1: K pair {2,3}
    const int srow  = lane >> 3;   // staging: sub-row within group of 4
    const int scol  = lane & 7;    // staging: b128 column

    v8f acc0 = {}, acc1 = {}, acc2 = {}, acc3 = {};
    float nrm = 0.f;

    // Stage 16 rows x 32 cols: each row is 128B = 8 lanes x b128, so one async
    // instruction covers 4 rows; 4 instructions fill a stage. Fully coalesced.
    auto issue_stage = [&](int ks, unsigned bufOff) {
      #pragma unroll
      for (int r = 0; r < 4; ++r) {
        int rr = r * 4 + srow;
        int row = row0 + rr; if (row >= N) row = N - 1;           // clamp tail
        const float* gp = patterns + (size_t)row * DIM + ks + scol * 4;
        unsigned ldst = bufOff + rr * ROW_B + scol * 16;
        asm volatile("global_load_async_to_lds_b128 %0, %1, off"
                     :: "v"(ldst), "v"((unsigned long long)(uintptr_t)gp)
                     : "memory");
      }
    };

    auto compute_stage = [&](int ks, const float* tileLds) {
      const float* arow = tileLds + m * ROW_F;
      #pragma unroll
      for (int kc = 0; kc < KCHUNK; kc += 16) {
        v2f a0 = *(const v2f*)(arow + kc + 0  + khalf * 2);
        v2f a1 = *(const v2f*)(arow + kc + 4  + khalf * 2);
        v2f a2 = *(const v2f*)(arow + kc + 8  + khalf * 2);
        v2f a3 = *(const v2f*)(arow + kc + 12 + khalf * 2);
        v2f b0 = *(const v2f*)(ldsx + ks + kc + 0  + khalf * 2);
        v2f b1 = *(const v2f*)(ldsx + ks + kc + 4  + khalf * 2);
        v2f b2 = *(const v2f*)(ldsx + ks + kc + 8  + khalf * 2);
        v2f b3 = *(const v2f*)(ldsx + ks + kc + 12 + khalf * 2);
        // ||p||^2 partial: lanes m and m+16 jointly cover all K of row m.
        nrm = fmaf(a0.x, a0.x, fmaf(a0.y, a0.y, nrm));
        nrm = fmaf(a1.x, a1.x, fmaf(a1.y, a1.y, nrm));
        nrm = fmaf(a2.x, a2.x, fmaf(a2.y, a2.y, nrm));
        nrm = fmaf(a3.x, a3.x, fmaf(a3.y, a3.y, nrm));
        // D = A(16x4) * B(4x16, x broadcast over N) + C ; 4 accs break RAW chain.
        acc0 = __builtin_amdgcn_wmma_f32_16x16x4_f32(false, a0, false, b0, (short)0, acc0, false, false);
        acc1 = __builtin_amdgcn_wmma_f32_16x16x4_f32(false, a1, false, b1, (short)0, acc1, false, false);
        acc2 = __builtin_amdgcn_wmma_f32_16x16x4_f32(false, a2, false, b2, (short)0, acc2, false, false);
        acc3 = __builtin_amdgcn_wmma_f32_16x16x4_f32(false, a3, false, b3, (short)0, acc3, false, false);
      }
    };

    // Software pipeline, last stage peeled -> branch-free steady state.
    issue_stage(0, buf0);
    int buf = 0;
    for (int ks = 0; ks < DIM - KCHUNK; ks += KCHUNK) {
      issue_stage(ks + KCHUNK, buf ? buf0 : buf1);
      // Async loads complete in order: <=4 outstanding means the current
      // stage's 4 loads have landed while the next 4 stay in flight.
      asm volatile("s_wait_asynccnt 4" ::: "memory");
      compute_stage(ks, buf ? tl1 : tl0);
      buf ^= 1;
    }
    asm volatile("s_wait_asynccnt 0" ::: "memory");
    compute_stage(DIM - KCHUNK, buf ? tl1 : tl0);

    v8f acc = acc0 + acc1 + acc2 + acc3;
    float nrm2 = nrm + __shfl_xor(nrm, 16);   // row-m norm^2 now in lanes m and m+16

    // Column N=0 of C/D holds the dots: lane 0 has M=0..7 (VGPR 0..7), lane 16 has M=8..15.
    float dot = 0.f;
    #pragma unroll
    for (int j = 0; j < 8; ++j) {
      float dlo = __shfl(acc[j], 0);
      float dhi = __shfl(acc[j], 16);
      if (lane == j)     dot = dlo;
      if (lane == j + 8) dot = dhi;
    }
    int rowIdx = row0 + lane;
    if (lane < 16 && rowIdx < N) {
      bestVal = dot / fmaxf(sqrtf(nrm2), 1e-8f);  // /||x|| is a shared scalar: argmax-invariant
      bestIdx = rowIdx;
    }
  }

  // Wave argmax (stable: prefer lower index on ties, matching first-max semantics).
  #pragma unroll
  for (int off = 16; off > 0; off >>= 1) {
    float ov = __shfl_xor(bestVal, off);
    int   oi = __shfl_xor(bestIdx, off);
    if (ov > bestVal || (ov == bestVal && oi < bestIdx)) { bestVal = ov; bestIdx = oi; }
  }
  if (lane == 0) { swv[wave] = bestVal; swi[wave] = bestIdx; }
  __syncthreads();
  if (tid == 0) {
    float bv = swv[0]; int bi = swi[0];
    #pragma unroll
    for (int w = 1; w < WAVES_PB; ++w)
      if (swv[w] > bv || (swv[w] == bv && swi[w] < bi)) { bv = swv[w]; bi = swi[w]; }
    ws[2 * blockIdx.x] = bv;
    ((int*)ws)[2 * blockIdx.x + 1] = bi;
  }
}

__global__ __launch_bounds__(512) void nd_phase2(const float* __restrict__ ws, int nblocks,
                                                 const float* __restrict__ patterns,
                                                 float* __restrict__ out, int d) {
  __shared__ float wv[16];
  __shared__ int   wi[16];
  __shared__ int   finalIdx;
  const int tid = threadIdx.x, lane = tid & 31, wave = tid >> 5;

  float bv = -FLT_MAX; int bi = 0x7fffffff;
  for (int i = tid; i < nblocks; i += 512) {
    float v = ws[2 * i]; int ii = ((const int*)ws)[2 * i + 1];
    if (v > bv || (v == bv && ii < bi)) { bv = v; bi = ii; }
  }
  #pragma unroll
  for (int off = 16; off > 0; off >>= 1) {
    float ov = __shfl_xor(bv, off);
    int   oi = __shfl_xor(bi, off);
    if (ov > bv || (ov == bv && oi < bi)) { bv = ov; bi = oi; }
  }
  if (lane == 0) { wv[wave] = bv; wi[wave] = bi; }
  __syncthreads();
  if (tid == 0) {
    float fv = wv[0]; int fi = wi[0];
    #pragma unroll
    for (int w = 1; w < 16; ++w)
      if (wv[w] > fv || (wv[w] == fv && wi[w] < fi)) { fv = wv[w]; fi = wi[w]; }
    finalIdx = fi;
  }
  __syncthreads();
  const int fi = finalIdx;
  for (int i = tid; i < d; i += 512) out[i] = patterns[(size_t)fi * d + i];
}

extern "C" void kernel_launch(void* const* d_in, const int* in_sizes, int n_in,
                              void* d_out, int out_size, void* d_ws, size_t ws_size,
                              hipStream_t stream) {
  const float* x        = (const float*)d_in[0];
  const float* patterns = (const float*)d_in[1];
  const int d = in_sizes[0];            // 512 (phase1 assumes DIM==512)
  const int N = in_sizes[1] / d;        // 100000
  float* out = (float*)d_out;
  float* ws  = (float*)d_ws;

  const int nTiles  = (N + TILE_ROWS - 1) / TILE_ROWS;
  const int nBlocks = (nTiles + WAVES_PB - 1) / WAVES_PB;

  nd_phase1<<<nBlocks, 256, 0, stream>>>(x, patterns, N, ws);
  nd_phase2<<<1, 512, 0, stream>>>(ws, nBlocks, patterns, out, d);
}